// BatchAllTripletLoss_36249523978553
// MI455X (gfx1250) — compile-verified
//
#include <hip/hip_runtime.h>
#include <hip/hip_bf16.h>
#include <math.h>

// ---------------------------------------------------------------------------
// BatchAllTripletLoss on gfx1250 (MI455X, wave32, WMMA)
//
//   d_in[0] : labels      int32  [512]
//   d_in[1] : embeddings  fp32   [512,128]
//   d_out   : fp32 [2] = { loss, frac_pos }
//
//   ws layout (floats):
//     sq    [512]        squared row norms (diag of Gram)
//     pdist [512*512]    Euclidean distance matrix
//     acc   [3]          { sum_tl, num_pos, num_valid }
// ---------------------------------------------------------------------------

#define B_N 512
#define D_N 128
#define MARGIN 0.2f

typedef __attribute__((ext_vector_type(2))) float v2f;
typedef __attribute__((ext_vector_type(8))) float v8f;

// --- Kernel 1: squared norms + zero accumulators ---------------------------
__global__ void tl_sqnorm_kernel(const float* __restrict__ emb,
                                 float* __restrict__ sq,
                                 float* __restrict__ acc) {
  int i = blockIdx.x * blockDim.x + threadIdx.x;   // 0..511
  if (i == 0) { acc[0] = 0.0f; acc[1] = 0.0f; acc[2] = 0.0f; }
  if (i < B_N) {
    const float4* row = (const float4*)(emb + i * D_N);
    float s = 0.0f;
#pragma unroll 8
    for (int k = 0; k < D_N / 4; ++k) {
      float4 v = row[k];
      s += v.x * v.x + v.y * v.y + v.z * v.z + v.w * v.w;
    }
    sq[i] = s;
  }
}

// --- Kernel 2: WMMA Gram-matrix + fused pdist epilogue ---------------------
// One wave32 computes one 16x16 tile of G = E * E^T via V_WMMA_F32_16X16X4_F32,
// then converts to Euclidean distances using the precomputed squared norms.
//
// f32 16x16x4 layouts (ISA 7.12.2):
//   A (16x4):  lane L, vgpr v -> (m = L%16, k = v + 2*(L/16))
//   B (4x16):  lane L, vgpr v -> (k = v + 2*(L/16), n = L%16)
//   C (16x16): lane L, vgpr v -> (m = v + 8*(L/16), n = L%16)
__global__ void tl_gram_pdist_kernel(const float* __restrict__ emb,
                                     const float* __restrict__ sq,
                                     float* __restrict__ pdist) {
  int gtid = blockIdx.x * blockDim.x + threadIdx.x;
  int wave = gtid >> 5;           // tile id, 0..1023  (32x32 tiles)
  int lane = threadIdx.x & 31;
  int ti = wave >> 5;             // tile row
  int tj = wave & 31;             // tile col
  int i0 = ti * 16;
  int j0 = tj * 16;

  int idx  = lane & 15;           // m for A-tile, n for B-tile
  int koff = (lane >> 4) * 2;     // 0 or 2

  const float* Abase = emb + (i0 + idx) * D_N + koff;   // anchor rows
  const float* Bbase = emb + (j0 + idx) * D_N + koff;   // other rows

  v8f c = {};                     // fp32 accumulator, 8 VGPRs
  for (int k = 0; k < D_N; k += 4) {
    v2f a = *(const v2f*)(Abase + k);   // (m, k+koff), (m, k+koff+1)
    v2f b = *(const v2f*)(Bbase + k);   // (k+koff, n), (k+koff+1, n)
    // D = A x B + C   (8 args: neg_a, A, neg_b, B, c_mod, C, reuse_a, reuse_b)
    c = __builtin_amdgcn_wmma_f32_16x16x4_f32(false, a, false, b,
                                              (short)0, c, false, false);
  }

  int col   = j0 + (lane & 15);
  int rbase = i0 + (lane >> 4) * 8;
  float sqc = sq[col];
#pragma unroll
  for (int v = 0; v < 8; ++v) {
    int row = rbase + v;
    float d2 = sq[row] + sqc - 2.0f * c[v];
    // sqrt safe at zero distance (matches reference subgradient handling)
    float d = (d2 > 0.0f) ? sqrtf(d2) : 0.0f;
    pdist[row * B_N + col] = d;
  }
}

// --- Kernel 3: per-anchor triplet reduction --------------------------------
// Block = anchor a. Positives: label[p]==label[a], p!=a. Negatives:
// label[n]!=label[a] (other distinctness constraints are implied).
// num_valid = |P_a| * |N_a| in closed form.
__global__ void tl_triplet_kernel(const int* __restrict__ labels,
                                  const float* __restrict__ pdist,
                                  float* __restrict__ acc) {
  __shared__ int   s_lab[B_N];
  __shared__ float s_d[B_N];
  __shared__ float s_r0[256];
  __shared__ float s_r1[256];

  int a   = blockIdx.x;
  int tid = threadIdx.x;

  const float* drow = pdist + a * B_N;
  for (int i = tid; i < B_N; i += 256) {
    s_lab[i] = labels[i];
    s_d[i]   = drow[i];
  }
  __syncthreads();

  int la = s_lab[a];

  float sum_tl = 0.0f;
  float n_pos  = 0.0f;
  for (int p = 0; p < B_N; ++p) {            // uniform branch per block
    if (s_lab[p] == la && p != a) {
      float cap = s_d[p] + MARGIN;           // d(a,p) + margin
      for (int n = tid; n < B_N; n += 256) { // 2 iters per thread
        if (s_lab[n] != la) {
          float t = cap - s_d[n];
          if (t > 0.0f)     sum_tl += t;
          if (t > 1e-16f)   n_pos  += 1.0f;
        }
      }
    }
  }

  // block reduction
  s_r0[tid] = sum_tl;
  s_r1[tid] = n_pos;
  __syncthreads();
  for (int s = 128; s > 0; s >>= 1) {
    if (tid < s) {
      s_r0[tid] += s_r0[tid + s];
      s_r1[tid] += s_r1[tid + s];
    }
    __syncthreads();
  }

  if (tid == 0) {
    atomicAdd(&acc[0], s_r0[0]);
    atomicAdd(&acc[1], s_r1[0]);
    // closed-form num_valid for this anchor
    int cntP = 0, cntN = 0;
    for (int i = 0; i < B_N; ++i) {
      if (s_lab[i] == la) { if (i != a) ++cntP; }
      else                { ++cntN; }
    }
    atomicAdd(&acc[2], (float)cntP * (float)cntN);
  }
}

// --- Kernel 4: finalize ----------------------------------------------------
__global__ void tl_finalize_kernel(const float* __restrict__ acc,
                                   float* __restrict__ out) {
  if (threadIdx.x == 0 && blockIdx.x == 0) {
    float sum_tl    = acc[0];
    float num_pos   = acc[1];
    float num_valid = acc[2];
    out[0] = sum_tl / (num_pos + 1e-16f);      // loss
    out[1] = num_pos / (num_valid + 1e-16f);   // frac_pos
  }
}

// ---------------------------------------------------------------------------
extern "C" void kernel_launch(void* const* d_in, const int* in_sizes, int n_in,
                              void* d_out, int out_size, void* d_ws, size_t ws_size,
                              hipStream_t stream) {
  const int*   labels = (const int*)d_in[0];
  const float* emb    = (const float*)d_in[1];
  float*       out    = (float*)d_out;

  float* sq    = (float*)d_ws;           // 512
  float* pdist = sq + B_N;               // 512*512
  float* acc   = pdist + B_N * B_N;      // 3

  // 1) squared norms + zero accumulators
  tl_sqnorm_kernel<<<2, 256, 0, stream>>>(emb, sq, acc);

  // 2) WMMA Gram + pdist: 1024 tiles, 8 waves/block -> 128 blocks
  tl_gram_pdist_kernel<<<128, 256, 0, stream>>>(emb, sq, pdist);

  // 3) triplet reduction: one block per anchor
  tl_triplet_kernel<<<B_N, 256, 0, stream>>>(labels, pdist, acc);

  // 4) finalize
  tl_finalize_kernel<<<1, 32, 0, stream>>>(acc, out);
}